// HungarianMatcher_76879914598787
// MI455X (gfx1250) — compile-verified
//
#include <hip/hip_runtime.h>

typedef __attribute__((ext_vector_type(2))) float v2f;
typedef __attribute__((ext_vector_type(8))) float v8f;

#define BS   16
#define QQ   300
#define CC   2
#define PP   320
#define NN   (BS * QQ)   // 4800 = 300 * 16
#define PS   96          // LDS stride per pred row (floats)
#define TS   112         // LDS stride per tgt row (floats)

// Pred row layout in LDS (stride PS):
//  [0:18)  x      [18:36) y     [36:54) x_abs  [54:72) y_abs
//  [72:92) id (17 + 3 zero pad) [92] sumsq(id) [93] prob0 [94] prob1
// Tgt row layout in LDS (stride TS):
//  [0:18)  tx     [18:36) ty    [36:54) tx_abs [54:72) ty_abs
//  [72:90) vis    [90:110) tid (17 + 3 zero pad)
//  [110] sumsq(tid)  [111] class id (as float)

__global__ __launch_bounds__(32)
void hungarian_cost_kernel(const float* __restrict__ pred_logits,
                           const float* __restrict__ pred_kpts,
                           const float* __restrict__ tgt_kpts,
                           const int*   __restrict__ tgt_ids,
                           float* __restrict__ out) {
    __shared__ float ldsP[16 * PS];
    __shared__ float ldsT[16 * TS];

    const int lane = threadIdx.x;       // 0..31 (wave32)
    const int n0 = blockIdx.x * 16;     // query-tile base (row of C)
    const int p0 = blockIdx.y * 16;     // target-tile base (col of C)

    // ---------------- Phase 1: per-row prep into LDS ----------------
    if (lane < 16) {
        const int m = lane;
        const int n = n0 + m;
        const float* kp = pred_kpts + n * 53;
        float* Pm = ldsP + m * PS;
        const float x0 = kp[0], y0 = kp[1];
        Pm[0] = x0;  Pm[18] = y0;  Pm[36] = x0;  Pm[54] = y0;
        float ss = 0.f;
        #pragma unroll
        for (int j = 1; j <= 17; ++j) {
            const float xj  = kp[3 * j - 1];      // out_k[:, 2::3]
            const float yj  = kp[3 * j];          // out_k[:, 3::3]
            const float idj = kp[3 * j + 1];      // out_k[:, 4::3]
            Pm[j]      = xj;
            Pm[18 + j] = yj;
            Pm[36 + j] = (xj - 0.5f) * 2.0f + x0; // x_abs
            Pm[54 + j] = (yj - 0.5f) * 2.0f + y0; // y_abs
            Pm[72 + (j - 1)] = idj;
            ss += idj * idj;
        }
        Pm[89] = 0.f; Pm[90] = 0.f; Pm[91] = 0.f; // K pad 17..19
        Pm[92] = ss;
        // 2-class softmax
        const float l0 = pred_logits[n * CC + 0];
        const float l1 = pred_logits[n * CC + 1];
        const float mx = fmaxf(l0, l1);
        const float e0 = expf(l0 - mx);
        const float e1 = expf(l1 - mx);
        const float inv = 1.0f / (e0 + e1);
        Pm[93] = e0 * inv;
        Pm[94] = e1 * inv;
    } else {
        const int r = lane - 16;
        const int p = p0 + r;
        const float* kt = tgt_kpts + p * 54;
        float* T = ldsT + r * TS;
        const float tx0 = kt[0], ty0 = kt[1];
        T[0] = tx0;  T[18] = ty0;  T[36] = tx0;  T[54] = ty0;
        T[72] = (kt[2] == 1.0f) ? 1.0f : 0.0f;    // vis[0]
        float ss = 0.f;
        #pragma unroll
        for (int j = 1; j <= 17; ++j) {
            const float txj = kt[3 * j];          // tgt[:, 0::3]
            const float tyj = kt[3 * j + 1];      // tgt[:, 1::3]
            const float slt = kt[3 * j + 2];      // tgt[:, 2::3] (vis) == tgt[:,5::3] (ids) slots
            T[j]      = txj;
            T[18 + j] = tyj;
            T[36 + j] = (txj - 0.5f) * 2.0f + tx0;
            T[54 + j] = (tyj - 0.5f) * 2.0f + ty0;
            T[72 + j] = (slt == 1.0f) ? 1.0f : 0.0f;
            T[90 + (j - 1)] = slt;                // tid[j-1] = tgt[:,5::3]
            ss += slt * slt;
        }
        T[107] = 0.f; T[108] = 0.f; T[109] = 0.f; // K pad 17..19
        T[110] = ss;
        T[111] = (float)tgt_ids[p];
    }

    __syncthreads();

    // ---------------- Phase 2: WMMA Gram tile (id dot products) ----------------
    // D[m,p] = sum_k id[m,k] * tid[p,k], K padded to 20 -> 5 steps of K=4.
    // f32 16x16x4 A fragment: lanes 0-15 M=lane K={kb,kb+1}; lanes 16-31 K={kb+2,kb+3}.
    const int colIdx = lane & 15;
    const int hi16   = lane >> 4;               // 0 or 1
    const int kh     = hi16 << 1;               // K sub-offset 0 or 2
    const float* Arow = ldsP + colIdx * PS + 72;
    const float* Brow = ldsT + colIdx * TS + 90;

    v8f c = {0.f, 0.f, 0.f, 0.f, 0.f, 0.f, 0.f, 0.f};
    #pragma unroll
    for (int s = 0; s < 5; ++s) {
        const int kb = 4 * s + kh;
        v2f a; a.x = Arow[kb]; a.y = Arow[kb + 1];
        v2f b; b.x = Brow[kb]; b.y = Brow[kb + 1];
        c = __builtin_amdgcn_wmma_f32_16x16x4_f32(
                false, a, false, b, (short)0, c, false, false);
    }

    float cd[8];
    #pragma unroll
    for (int r = 0; r < 8; ++r) cd[r] = c[r];

    // ---------------- Phase 3: finish costs per C-tile element ----------------
    // C layout: lane 0-15 -> N=lane, VGPR r -> M=r ; lane 16-31 -> M=r+8.
    const float* T = ldsT + colIdx * TS;
    const float st    = T[110];
    const int   tcls  = (T[111] > 0.5f) ? 1 : 0;
    const float v0    = T[72];
    const float tx0   = T[0];
    const float ty0   = T[18];

    float acc[8];
    #pragma unroll
    for (int r = 0; r < 8; ++r) {
        const float* Pm = ldsP + (r + 8 * hi16) * PS;
        // kpts_class: sqrt(|a|^2 + |b|^2 - 2 a.b)
        const float d2 = Pm[92] + st - 2.0f * cd[r];
        const float ckc = sqrtf(fmaxf(d2, 0.0f));
        // class: -prob[n, tgt_id[p]]
        const float prob = tcls ? Pm[94] : Pm[93];
        // ctrs: 2-D distance on (x0,y0) with vis[0]
        const float dx = v0 * (Pm[0]  - tx0);
        const float dy = v0 * (Pm[18] - ty0);
        acc[r] = ckc - prob + sqrtf(dx * dx + dy * dy);
    }

    // deltas + kpts: L1 over joints 1..17, relative + absolute coords
    for (int j = 1; j <= 17; ++j) {
        const float vj  = T[72 + j];
        const float tx  = T[j];
        const float ty  = T[18 + j];
        const float txa = T[36 + j];
        const float tya = T[54 + j];
        #pragma unroll
        for (int r = 0; r < 8; ++r) {
            const float* Pm = ldsP + (r + 8 * hi16) * PS;
            acc[r] += vj * (fabsf(Pm[j]      - tx)  + fabsf(Pm[18 + j] - ty) +
                            fabsf(Pm[36 + j] - txa) + fabsf(Pm[54 + j] - tya));
        }
    }

    #pragma unroll
    for (int r = 0; r < 8; ++r) {
        const int n = n0 + r + 8 * hi16;
        out[n * PP + p0 + colIdx] = acc[r];   // lanes 0-15: consecutive p -> coalesced
    }
}

extern "C" void kernel_launch(void* const* d_in, const int* in_sizes, int n_in,
                              void* d_out, int out_size, void* d_ws, size_t ws_size,
                              hipStream_t stream) {
    const float* pred_logits = (const float*)d_in[0];
    const float* pred_kpts   = (const float*)d_in[1];
    const float* tgt_kpts    = (const float*)d_in[2];
    const int*   tgt_ids     = (const int*)d_in[3];
    float* out = (float*)d_out;

    dim3 grid(NN / 16, PP / 16);   // 300 x 20 tiles, one wave32 per 16x16 tile
    hungarian_cost_kernel<<<grid, dim3(32), 0, stream>>>(
        pred_logits, pred_kpts, tgt_kpts, tgt_ids, out);
}